// lut_kernel_49538152792187
// MI455X (gfx1250) — compile-verified
//
#include <hip/hip_runtime.h>
#include <cstdint>
#include <cstddef>

#define LUT_K    6
#define OUT_NUM  1024
#define LUT_NUM  64
#define BR       32                    // 16 * 2 batch rows
#define FEAT     (LUT_NUM * LUT_K)     // 384
#define TBL      (1 << LUT_K)          // 64
#define WS_STRIDE 66                   // padded row stride for bank-conflict-free reads
#define BLOCK    256                   // 8 waves of 32

typedef int v4i  __attribute__((vector_size(16)));
typedef unsigned int u32x4 __attribute__((ext_vector_type(4)));
typedef int i32x4 __attribute__((ext_vector_type(4)));
typedef int i32x8 __attribute__((ext_vector_type(8)));
typedef __attribute__((address_space(3))) char lds_char;

// wq = (tanh(w)+1)/2 == sigmoid(2w) = 1 / (1 + exp2(-2*log2(e) * w))
__device__ __forceinline__ float soft_bit(float w) {
    float e = __builtin_amdgcn_exp2f(w * -2.8853900817779268f);
    return __builtin_amdgcn_rcpf(1.0f + e);
}

__global__ __launch_bounds__(BLOCK) void lut_tree_kernel(
    const float* __restrict__ x,   // [BR][OUT_NUM][FEAT]
    const float* __restrict__ w,   // [OUT_NUM][LUT_NUM][TBL]
    float* __restrict__ y)         // [BR][OUT_NUM][LUT_NUM]
{
    __shared__ __align__(16) float xs[BR * FEAT];            // 48 KB
    __shared__ __align__(16) float ws[LUT_NUM * WS_STRIDE];  // 16.5 KB

    const int tid  = threadIdx.x;
    const int out  = blockIdx.x;
    const int lane = tid & 31;
    const int wave = tid >> 5;

    // ---------------- stage x tile [BR x FEAT] into LDS ----------------
    // element layout: x[(br*OUT_NUM + out)*FEAT + f]
    const float* xg = x + (size_t)out * FEAT;

#if __has_builtin(__builtin_amdgcn_tensor_load_to_lds) && __has_builtin(__builtin_amdgcn_s_wait_tensorcnt)
    // One TDM 2D-tile load: 32 rows x 1536B, row stride = OUT_NUM*FEAT elements.
    // Issued by wave 0 only (TDM ignores EXEC, so use a wave-uniform branch).
    if (__builtin_amdgcn_readfirstlane(tid) == 0) {
        const uint64_t ga  = (uint64_t)(uintptr_t)xg;                 // tile start (bytes)
        const uint32_t lds = (uint32_t)(uintptr_t)(lds_char*)(void*)xs; // LDS byte address

        u32x4 g0;
        g0[0] = 1u;                                   // count=1, user mode, no gather
        g0[1] = lds;                                  // lds_addr [63:32]
        g0[2] = (uint32_t)(ga & 0xFFFFFFFFu);         // global_addr [95:64]
        g0[3] = (uint32_t)((ga >> 32) & 0x01FFFFFFu)  // global_addr [120:96]
              | 0x80000000u;                          // type=2 ("image") [127:126]

        i32x8 g1;
        g1[0] = 0x00020000;            // workgroup_mask=0, data_size=2 (4B), no flags
        g1[1] = (FEAT & 0xFFFF) << 16; // atomic_barrier_addr=0 | tensor_dim0 lo16 (384)
        g1[2] = (BR & 0xFFFF) << 16;   // tensor_dim0 hi16 (0) | tensor_dim1 lo16 (32)
        g1[3] = (FEAT & 0xFFFF) << 16; // tensor_dim1 hi16 (0) | tile_dim0 (384)
        g1[4] = BR;                    // tile_dim1 (32) | tile_dim2=0
        g1[5] = OUT_NUM * FEAT;        // tensor_dim0_stride lo32 (393216)
        g1[6] = 0;                     // tensor_dim0_stride hi16 | dim1_stride lo16
        g1[7] = 0;                     // tensor_dim1_stride (unused for 2D)

        i32x4 g2 = {0, 0, 0, 0};       // dims 2/3 unused (2D tile)
        i32x4 g3 = {0, 0, 0, 0};
        i32x8 g4 = {0, 0, 0, 0, 0, 0, 0, 0};  // extra group (clang-23 form): zero-filled

        __builtin_amdgcn_tensor_load_to_lds(g0, g1, g2, g3, g4, /*cpol=*/0);
    }
#elif __has_builtin(__builtin_amdgcn_global_load_async_to_lds_b128)
    // 12 async b128 copies per thread, division-free index math:
    // each wave owns 4 rows (br), 96 float4 per row covered by 32 lanes x 3.
    #pragma unroll
    for (int r = 0; r < 4; ++r) {
        #pragma unroll
        for (int c = 0; c < 3; ++c) {
            const int br = wave * 4 + r;
            const int f4 = lane + 32 * c;
            const float* src = xg + (size_t)br * (OUT_NUM * FEAT) + f4 * 4;
            float*       dst = &xs[br * FEAT + f4 * 4];
            __builtin_amdgcn_global_load_async_to_lds_b128((v4i*)src, (v4i*)dst, 0, 0);
        }
    }
#else
    #pragma unroll
    for (int r = 0; r < 4; ++r) {
        #pragma unroll
        for (int c = 0; c < 3; ++c) {
            const int br = wave * 4 + r;
            const int f4 = lane + 32 * c;
            const float4 s = *(const float4*)(xg + (size_t)br * (OUT_NUM * FEAT) + f4 * 4);
            *(float4*)&xs[br * FEAT + f4 * 4] = s;
        }
    }
#endif

    // ---------------- stage w slice, apply sigmoid(2w) (overlaps the DMA) ----
    const float4* wg4 = (const float4*)(w + (size_t)out * (LUT_NUM * TBL));
    #pragma unroll
    for (int i = 0; i < 4; ++i) {
        int q = tid + BLOCK * i;         // float4 index 0..1023
        float4 v = wg4[q];
        int lut = q >> 4;                // 16 float4 per lut row
        int idx = (q & 15) * 4;
        float* dst = &ws[lut * WS_STRIDE + idx];
        dst[0] = soft_bit(v.x);
        dst[1] = soft_bit(v.y);
        dst[2] = soft_bit(v.z);
        dst[3] = soft_bit(v.w);
    }

#if __has_builtin(__builtin_amdgcn_tensor_load_to_lds) && __has_builtin(__builtin_amdgcn_s_wait_tensorcnt)
    if (__builtin_amdgcn_readfirstlane(tid) == 0) {
        __builtin_amdgcn_s_wait_tensorcnt(0);   // TENSORcnt is per-wave: issuer waits
    }
#elif __has_builtin(__builtin_amdgcn_global_load_async_to_lds_b128)
    __builtin_amdgcn_s_wait_asynccnt(0);
#endif
    __syncthreads();

    // ---------------- compute ----------------
    // lut = tid & 63 -> lanes within a wave map to consecutive luts
    // (x-tile LDS reads stride 6 dwords across lanes: conflict-free;
    //  output stores are coalesced)
    const int lut = tid & 63;
    const int brb = (tid >> 6) * 8;      // each thread handles 8 batch rows

    // table in registers, reused for all 8 br iterations
    float2 t[32];
    #pragma unroll
    for (int i = 0; i < 32; ++i)
        t[i] = *(const float2*)&ws[lut * WS_STRIDE + 2 * i];

    float* yo = y + (size_t)out * LUT_NUM + lut;

    #pragma unroll
    for (int j = 0; j < 8; ++j) {
        const int br = brb + j;
        const float* bx = &xs[br * FEAT + lut * LUT_K];
        const float b0 = bx[0], b1 = bx[1], b2 = bx[2];
        const float b3 = bx[3], b4 = bx[4], b5 = bx[5];

        float v[32];
        #pragma unroll
        for (int i = 0; i < 32; ++i) v[i] = fmaf(b0, t[i].y - t[i].x, t[i].x);
        #pragma unroll
        for (int i = 0; i < 16; ++i) v[i] = fmaf(b1, v[2*i+1] - v[2*i], v[2*i]);
        #pragma unroll
        for (int i = 0; i < 8;  ++i) v[i] = fmaf(b2, v[2*i+1] - v[2*i], v[2*i]);
        #pragma unroll
        for (int i = 0; i < 4;  ++i) v[i] = fmaf(b3, v[2*i+1] - v[2*i], v[2*i]);
        #pragma unroll
        for (int i = 0; i < 2;  ++i) v[i] = fmaf(b4, v[2*i+1] - v[2*i], v[2*i]);
        const float r = fmaf(b5, v[1] - v[0], v[0]);

        yo[(size_t)br * (OUT_NUM * LUT_NUM)] = r;
    }
}

extern "C" void kernel_launch(void* const* d_in, const int* in_sizes, int n_in,
                              void* d_out, int out_size, void* d_ws, size_t ws_size,
                              hipStream_t stream) {
    (void)in_sizes; (void)n_in; (void)out_size; (void)d_ws; (void)ws_size;
    const float* x = (const float*)d_in[0];
    const float* w = (const float*)d_in[1];
    float* y = (float*)d_out;
    lut_tree_kernel<<<OUT_NUM, BLOCK, 0, stream>>>(x, w, y);
}